// HashEncoder_2207613190663
// MI455X (gfx1250) — compile-verified
//
#include <hip/hip_runtime.h>

// ---------------------------------------------------------------------------
// Instant-NGP hash-grid encoder for MI455X (gfx1250, wave32).
//
// Layout: thread t -> point = t>>4, level = t&15.
//   - 16 consecutive lanes = 16 levels of one point  -> output store of the
//     point's 32 floats is one contiguous 128B region per half-wave.
//   - Each thread: 8 independent b64 gathers (float2 corner features),
//     trilinear weights, 2-channel FMA reduction, NT b64 store.
//
// Cache policy: embeddings (48.8 MB) are L2-resident (192 MB L2); output
// (128 MB) is stored non-temporally so it cannot evict the tables.
//
// Per-level constants packed {res, offset} -> single b64 load per thread.
// ---------------------------------------------------------------------------

typedef __attribute__((ext_vector_type(2))) float v2f;
typedef __attribute__((ext_vector_type(2))) unsigned int v2u;

#define NLEVELS 16
#define TMASK   0x7FFFFu          // T - 1, T = 2^19
#define P1      2654435761u
#define P2      805459861u

// {RES[i], OFFSET[i]} ; RES[i] = ceil(16 * 2^(7i/15)), OFFSET = cumsum(min(res^3, 2^19))
// Levels >= 5 hashed (81^3 = 531441 > 524288).
__constant__ v2u c_lvl[NLEVELS] = {
    {16u,   0u},       {23u,   4096u},    {31u,   16263u},   {43u,   46054u},
    {59u,   125561u},  {81u,   330940u},  {112u,  855228u},  {154u,  1379516u},
    {213u,  1903804u}, {295u,  2428092u}, {407u,  2952380u}, {562u,  3476668u},
    {777u,  4000956u}, {1073u, 4525244u}, {1483u, 5049532u}, {2048u, 5573820u}};

__global__ __launch_bounds__(256) void hashgrid_encode_kernel(
    const float* __restrict__ x,          // [B,3]
    const v2f*   __restrict__ emb,        // [TOTAL_ROWS] of float2
    const int*   __restrict__ act_levels, // scalar
    float*       __restrict__ out,        // [B,32]
    int n_point_level)                    // B * 16
{
    const int t = blockIdx.x * 256 + threadIdx.x;
    if (t >= n_point_level) return;

    const int point = t >> 4;
    const int level = t & 15;

    // ---- per-level constants: one b64 load ----
    const v2u lvl = c_lvl[level];
    const unsigned int res  = lvl.x;
    const unsigned int base = lvl.y;
    const bool hashed = (level >= 5);

    // ---- point position (broadcast across the 16 lanes of this point) ----
    const float x0 = x[point * 3 + 0];
    const float x1 = x[point * 3 + 1];
    const float x2 = x[point * 3 + 2];
    const float u0 = (x0 + 1.0f) * 0.5f;
    const float u1 = (x1 + 1.0f) * 0.5f;
    const float u2 = (x2 + 1.0f) * 0.5f;

    const float rm1 = (float)(res - 1u);
    const float rm2 = (float)(res - 2u);

    const float p0 = u0 * rm1, p1 = u1 * rm1, p2 = u2 * rm1;
    const float q0 = fminf(fmaxf(floorf(p0), 0.0f), rm2);
    const float q1 = fminf(fmaxf(floorf(p1), 0.0f), rm2);
    const float q2 = fminf(fmaxf(floorf(p2), 0.0f), rm2);
    const float fr0 = p0 - q0, fr1 = p1 - q1, fr2 = p2 - q2;
    const unsigned int i0 = (unsigned int)q0;
    const unsigned int i1 = (unsigned int)q1;
    const unsigned int i2 = (unsigned int)q2;

    // ---- 8 corner indices (branch-free hashed/dense select) ----
    unsigned int idx[8];
#pragma unroll
    for (int k = 0; k < 8; ++k) {
        const unsigned int cx = i0 + (unsigned int)(k & 1);
        const unsigned int cy = i1 + (unsigned int)((k >> 1) & 1);
        const unsigned int cz = i2 + (unsigned int)((k >> 2) & 1);
        const unsigned int ih = (cx ^ (cy * P1) ^ (cz * P2)) & TMASK;
        const unsigned int id = cx + res * (cy + res * cz);
        idx[k] = base + (hashed ? ih : id);
    }

    // ---- issue all 8 gathers before any use (deep MLP) ----
    v2f g[8];
#pragma unroll
    for (int k = 0; k < 8; ++k) {
        g[k] = emb[idx[k]];               // global_load_b64 gather, L2-resident
    }

    // ---- trilinear weights + 2-channel FMA reduction ----
    float a0 = 0.0f, a1 = 0.0f;
#pragma unroll
    for (int k = 0; k < 8; ++k) {
        const float wx = (k & 1)        ? fr0 : 1.0f - fr0;
        const float wy = ((k >> 1) & 1) ? fr1 : 1.0f - fr1;
        const float wz = ((k >> 2) & 1) ? fr2 : 1.0f - fr2;
        const float w = wx * wy * wz;
        a0 = fmaf(w, g[k].x, a0);
        a1 = fmaf(w, g[k].y, a1);
    }

    // ---- active_levels mask (uniform scalar load) ----
    const int al = *act_levels;
    if (level >= al) { a0 = 0.0f; a1 = 0.0f; }

    // ---- non-temporal b64 store: keep the 128MB output stream out of L2 ----
    v2f r; r.x = a0; r.y = a1;
    v2f* dst = (v2f*)(out + (size_t)point * 32 + level * 2);
    __builtin_nontemporal_store(r, dst);
}

extern "C" void kernel_launch(void* const* d_in, const int* in_sizes, int n_in,
                              void* d_out, int out_size, void* d_ws, size_t ws_size,
                              hipStream_t stream) {
    const float* x   = (const float*)d_in[0];   // [B,3] float32
    const v2f*   emb = (const v2f*)d_in[1];     // [TOTAL_ROWS,2] float32
    const int*   act = (const int*)d_in[2];     // scalar int

    const int B = in_sizes[0] / 3;
    const int n_point_level = B * NLEVELS;

    const int block = 256;
    const int grid  = (n_point_level + block - 1) / block;

    hashgrid_encode_kernel<<<grid, block, 0, stream>>>(
        x, emb, act, (float*)d_out, n_point_level);
}